// CustomStellarEncoder_1_51342039056841
// MI455X (gfx1250) — compile-verified
//
#include <hip/hip_runtime.h>

typedef __attribute__((ext_vector_type(2))) float v2f;
typedef __attribute__((ext_vector_type(8))) float v8f;

#define HID 128
#define DIN 256
#define EPS 1e-5f

// ---------------- zero workspace ----------------
__global__ void k_zero(float* __restrict__ p, long n) {
    long i = (long)blockIdx.x * blockDim.x + threadIdx.x;
    long stride = (long)gridDim.x * blockDim.x;
    for (; i < n; i += stride) p[i] = 0.0f;
}

// ---------------- GEMM1: z = x@W1 + b1, accumulate BN1 stats ----------------
// one wave computes one 16x16 tile; block = 8 waves = 8 column tiles (128 cols)
__global__ __launch_bounds__(256)
void k_gemm1(const float* __restrict__ x, const float* __restrict__ W1,
             const float* __restrict__ b1, float* __restrict__ z,
             float* __restrict__ sum1, float* __restrict__ sq1, int N)
{
    __shared__ float As[16 * DIN];              // 16 KB x-tile, shared by 8 waves
    const int m0  = blockIdx.x * 16;
    const int tid = threadIdx.x;

    for (int idx = tid; idx < 16 * DIN; idx += 256) {
        int r = idx >> 8, c = idx & (DIN - 1);
        int row = m0 + r;
        As[idx] = (row < N) ? x[(long)row * DIN + c] : 0.0f;
    }
    __syncthreads();

    const int lane  = tid & 31;
    const int n0    = (tid >> 5) * 16;          // wave -> column tile
    const int m     = lane & 15;
    const int khalf = lane >> 4;                // 0 or 1
    const int n     = lane & 15;

    v8f c = {};
    #pragma unroll 8
    for (int k0 = 0; k0 < DIN; k0 += 4) {
        int kk = k0 + 2 * khalf;
        v2f a, b;
        a[0] = As[m * DIN + kk];
        a[1] = As[m * DIN + kk + 1];
        b[0] = W1[kk * HID + n0 + n];
        b[1] = W1[(kk + 1) * HID + n0 + n];
        c = __builtin_amdgcn_wmma_f32_16x16x4_f32(false, a, false, b,
                                                  (short)0, c, false, false);
    }

    const float bias = b1[n0 + n];
    float s = 0.0f, q = 0.0f;
    #pragma unroll
    for (int v = 0; v < 8; ++v) {
        int row = m0 + v + 8 * khalf;
        if (row < N) {
            float val = c[v] + bias;
            z[(long)row * HID + n0 + n] = val;
            s += val;
            q += val * val;
        }
    }
    atomicAdd(&sum1[n0 + n], s);
    atomicAdd(&sq1[n0 + n], q);
}

// ---------------- BN finalize: mu, rsigma ----------------
__global__ void k_bnfin(const float* __restrict__ sum, const float* __restrict__ sq,
                        float* __restrict__ mu, float* __restrict__ rs, float invN)
{
    int i = threadIdx.x;                        // 128 threads
    float m = sum[i] * invN;
    float var = fmaxf(sq[i] * invN - m * m, 0.0f);
    mu[i] = m;
    rs[i] = rsqrtf(var + EPS);
}

// ---------------- BN apply (optionally ReLU), in place ----------------
__global__ __launch_bounds__(256)
void k_bnapply(float* __restrict__ z, const float* __restrict__ mu,
               const float* __restrict__ rs, const float* __restrict__ g,
               const float* __restrict__ be, long total, int relu)
{
    long i = (long)blockIdx.x * blockDim.x + threadIdx.x;
    long stride = (long)gridDim.x * blockDim.x;
    for (; i < total; i += stride) {
        int ch = (int)(i & (HID - 1));
        float v = (z[i] - mu[ch]) * rs[ch] * g[ch] + be[ch];
        z[i] = relu ? fmaxf(v, 0.0f) : v;
    }
}

// ---------------- edge scatter: msg[dst] += feat[src]; deg[dst] += 1 ----------------
// one wave (32 lanes) per edge; float4 per lane covers all 128 channels
__global__ __launch_bounds__(256)
void k_scatter(const int* __restrict__ ei, const float* __restrict__ feat,
               float* __restrict__ msg, float* __restrict__ deg, int E)
{
    int e = blockIdx.x * 8 + (threadIdx.x >> 5);
    if (e >= E) return;
    int lane = threadIdx.x & 31;
    int src = ei[e];
    int dst = ei[E + e];
    const float4* fp = (const float4*)(feat + (long)src * HID);
    float4 v = fp[lane];
    float* mp = msg + (long)dst * HID + lane * 4;
    atomicAdd(mp + 0, v.x);
    atomicAdd(mp + 1, v.y);
    atomicAdd(mp + 2, v.z);
    atomicAdd(mp + 3, v.w);
    if (lane == 0) atomicAdd(deg + dst, 1.0f);
}

// ---------------- GEMM2: out = mean@Wl + bl + feat@Wr, accumulate BN2 stats ----------------
__global__ __launch_bounds__(256)
void k_gemm2(const float* __restrict__ msg, const float* __restrict__ deg,
             const float* __restrict__ feat, const float* __restrict__ Wl,
             const float* __restrict__ bl, const float* __restrict__ Wr,
             float* __restrict__ out2, float* __restrict__ sum2,
             float* __restrict__ sq2, int N)
{
    __shared__ float Ms[16 * HID];              // mean tile
    __shared__ float Fs[16 * HID];              // feat tile
    __shared__ float sinv[16];
    const int m0  = blockIdx.x * 16;
    const int tid = threadIdx.x;

    if (tid < 16) {
        int row = m0 + tid;
        sinv[tid] = (row < N) ? 1.0f / fmaxf(deg[row], 1.0f) : 0.0f;
    }
    __syncthreads();

    for (int idx = tid; idx < 16 * HID; idx += 256) {
        int r = idx >> 7;
        int row = m0 + r;
        if (row < N) {
            long g = (long)row * HID + (idx & (HID - 1));
            Ms[idx] = msg[g] * sinv[r];
            Fs[idx] = feat[g];
        } else {
            Ms[idx] = 0.0f;
            Fs[idx] = 0.0f;
        }
    }
    __syncthreads();

    const int lane  = tid & 31;
    const int n0    = (tid >> 5) * 16;
    const int m     = lane & 15;
    const int khalf = lane >> 4;
    const int n     = lane & 15;

    v8f c = {};
    #pragma unroll 8
    for (int k0 = 0; k0 < HID; k0 += 4) {       // mean @ Wl
        int kk = k0 + 2 * khalf;
        v2f a, b;
        a[0] = Ms[m * HID + kk];
        a[1] = Ms[m * HID + kk + 1];
        b[0] = Wl[kk * HID + n0 + n];
        b[1] = Wl[(kk + 1) * HID + n0 + n];
        c = __builtin_amdgcn_wmma_f32_16x16x4_f32(false, a, false, b,
                                                  (short)0, c, false, false);
    }
    #pragma unroll 8
    for (int k0 = 0; k0 < HID; k0 += 4) {       // + feat @ Wr
        int kk = k0 + 2 * khalf;
        v2f a, b;
        a[0] = Fs[m * HID + kk];
        a[1] = Fs[m * HID + kk + 1];
        b[0] = Wr[kk * HID + n0 + n];
        b[1] = Wr[(kk + 1) * HID + n0 + n];
        c = __builtin_amdgcn_wmma_f32_16x16x4_f32(false, a, false, b,
                                                  (short)0, c, false, false);
    }

    const float bias = bl[n0 + n];
    float s = 0.0f, q = 0.0f;
    #pragma unroll
    for (int v = 0; v < 8; ++v) {
        int row = m0 + v + 8 * khalf;
        if (row < N) {
            float val = c[v] + bias;
            out2[(long)row * HID + n0 + n] = val;
            s += val;
            q += val * val;
        }
    }
    atomicAdd(&sum2[n0 + n], s);
    atomicAdd(&sq2[n0 + n], q);
}

extern "C" void kernel_launch(void* const* d_in, const int* in_sizes, int n_in,
                              void* d_out, int out_size, void* d_ws, size_t ws_size,
                              hipStream_t stream) {
    const float* x   = (const float*)d_in[0];
    const int*   ei  = (const int*)  d_in[1];
    const float* W1  = (const float*)d_in[2];
    const float* b1  = (const float*)d_in[3];
    const float* g1  = (const float*)d_in[4];
    const float* be1 = (const float*)d_in[5];
    const float* Wl  = (const float*)d_in[6];
    const float* bl  = (const float*)d_in[7];
    const float* Wr  = (const float*)d_in[8];
    const float* g2  = (const float*)d_in[9];
    const float* be2 = (const float*)d_in[10];

    const int N = in_sizes[0] / DIN;
    const int E = in_sizes[1] / 2;

    // workspace layout (floats): msg[N*128] | deg[N] | stats/bn params [1024]
    float* ws   = (float*)d_ws;
    float* msg  = ws;
    float* deg  = ws + (long)N * HID;
    float* st   = deg + N;
    float* sum1 = st;        float* sq1 = st + 128;
    float* sum2 = st + 256;  float* sq2 = st + 384;
    float* mu1  = st + 512;  float* rs1 = st + 640;
    float* mu2  = st + 768;  float* rs2 = st + 896;

    float* feat = (float*)d_out;                 // output 0: [N, HID]
    float* out2 = (float*)d_out + (long)N * HID; // output 1: [N, HID]

    const long zn  = (long)N * HID + N + 1024;
    const long tot = (long)N * HID;
    const int mtiles = (N + 15) / 16;

    k_zero<<<2048, 256, 0, stream>>>(ws, zn);
    k_gemm1<<<mtiles, 256, 0, stream>>>(x, W1, b1, feat, sum1, sq1, N);
    k_bnfin<<<1, 128, 0, stream>>>(sum1, sq1, mu1, rs1, 1.0f / (float)N);
    k_bnapply<<<2048, 256, 0, stream>>>(feat, mu1, rs1, g1, be1, tot, 1);
    k_scatter<<<(E + 7) / 8, 256, 0, stream>>>(ei, feat, msg, deg, E);
    k_gemm2<<<mtiles, 256, 0, stream>>>(msg, deg, feat, Wl, bl, Wr, out2, sum2, sq2, N);
    k_bnfin<<<1, 128, 0, stream>>>(sum2, sq2, mu2, rs2, 1.0f / (float)N);
    k_bnapply<<<2048, 256, 0, stream>>>(out2, mu2, rs2, g2, be2, tot, 0);
}